// PaiConv_77429670412665
// MI455X (gfx1250) — compile-verified
//
#include <hip/hip_runtime.h>
#include <math.h>

#define B_   8
#define N_   4096
#define C_   64
#define K_   20
#define KS_  16
#define OC_  64
#define TILE 16

typedef __attribute__((ext_vector_type(2))) float v2f;
typedef __attribute__((ext_vector_type(8))) float v8f;

// ---------------------------------------------------------------------------
// Kernel 1: KNN (top-20 by -squared-distance, stable, self first)
// ---------------------------------------------------------------------------
__global__ void knn_kernel(const float* __restrict__ x, int* __restrict__ idxOut) {
    extern __shared__ float sX[];                       // [3][N_] = 48 KB
    const int blocksPerBatch = N_ / 256;
    const int b = blockIdx.x / blocksPerBatch;
    const int nbase = (blockIdx.x % blocksPerBatch) * 256;
    const float* xb = x + (size_t)b * 3 * N_;
    for (int t = threadIdx.x; t < 3 * N_; t += blockDim.x) sX[t] = xb[t];
    __syncthreads();

    const int n = nbase + threadIdx.x;
    const float xn0 = sX[0 * N_ + n], xn1 = sX[1 * N_ + n], xn2 = sX[2 * N_ + n];
    const float xxn = xn0 * xn0 + xn1 * xn1 + xn2 * xn2;

    float bs[K_]; int bi[K_];
#pragma unroll
    for (int j = 0; j < K_; ++j) { bs[j] = -3.0e38f; bi[j] = 0; }

    for (int m = 0; m < N_; ++m) {
        float m0 = sX[m], m1 = sX[N_ + m], m2 = sX[2 * N_ + m];
        float dot = xn0 * m0 + xn1 * m1 + xn2 * m2;
        float xxm = m0 * m0 + m1 * m1 + m2 * m2;
        float s = 2.0f * dot - xxn - xxm;               // = -dist^2, larger = nearer
        if (s > bs[K_ - 1]) {                           // guarded: rare after warmup
            int id = m;
#pragma unroll
            for (int j = 0; j < K_; ++j) {              // stable descending insert
                bool better = s > bs[j];
                float tf = bs[j]; int ti = bi[j];
                if (better) { bs[j] = s; bi[j] = id; s = tf; id = ti; }
            }
        }
    }
    int* o = idxOut + ((size_t)b * N_ + n) * K_;
#pragma unroll
    for (int j = 0; j < K_; ++j) o[j] = bi[j];
}

// ---------------------------------------------------------------------------
// Kernel 2: fused feats7 / aw / aggregation / WMMA conv GEMM (pre-BN output)
// LDS layout (float units):
//   sIdx  @ 0      (320 ints)     sXsp @ 320   (960)
//   sF7   @ 1280   (2560)         sAw  @ 3840  (5120)
//   sKern @ 8960   (48)           sMW  @ 9008  (448)
//   sMB   @ 9456   (64)           sAgg @ 9520  (32768)   total 42288 f = 169152 B
// ---------------------------------------------------------------------------
__global__ void paiconv_kernel(const float* __restrict__ x,
                               const float* __restrict__ feature,
                               const float* __restrict__ kern,
                               const float* __restrict__ mlp_w,
                               const float* __restrict__ mlp_b,
                               const float* __restrict__ conv_w,
                               const float* __restrict__ conv_b,
                               const int* __restrict__ idxIn,
                               float* __restrict__ outRaw) {
    extern __shared__ float sm[];
    int*   sIdx  = (int*)sm;
    float* sXsp  = sm + 320;
    float* sF7   = sm + 1280;
    float* sAw   = sm + 3840;
    float* sKern = sm + 8960;
    float* sMW   = sm + 9008;
    float* sMB   = sm + 9456;
    float* sAgg  = sm + 9520;

    const int tilesPerBatch = N_ / TILE;                // 256
    const int b  = blockIdx.x / tilesPerBatch;
    const int n0 = (blockIdx.x % tilesPerBatch) * TILE;
    const int tid = threadIdx.x;                        // 128 threads = 4 waves

    for (int t = tid; t < 48;  t += 128) sKern[t] = kern[t];      // [3][16]
    for (int t = tid; t < 448; t += 128) sMW[t]   = mlp_w[t];     // [64][7]
    for (int t = tid; t < 64;  t += 128) sMB[t]   = mlp_b[t];
    for (int t = tid; t < TILE * K_; t += 128)
        sIdx[t] = idxIn[((size_t)b * N_ + n0) * K_ + t];
    __syncthreads();

    // gather neighbor coords x[b, :, m]
    const float* xb = x + (size_t)b * 3 * N_;
    for (int t = tid; t < TILE * K_; t += 128) {
        int m = sIdx[t];
        sXsp[t * 3 + 0] = xb[m];
        sXsp[t * 3 + 1] = xb[N_ + m];
        sXsp[t * 3 + 2] = xb[2 * N_ + m];
    }
    __syncthreads();

    // feats7 = [center(3), rel(3), dis, pad]
    for (int t = tid; t < TILE * K_; t += 128) {
        int p = t / K_;
        float cx = sXsp[(p * K_) * 3 + 0];
        float cy = sXsp[(p * K_) * 3 + 1];
        float cz = sXsp[(p * K_) * 3 + 2];
        float rx = sXsp[t * 3 + 0] - cx;
        float ry = sXsp[t * 3 + 1] - cy;
        float rz = sXsp[t * 3 + 2] - cz;
        float dis = sqrtf(rx * rx + ry * ry + rz * rz);
        float* f7 = sF7 + t * 8;
        f7[0] = cx; f7[1] = cy; f7[2] = cz;
        f7[3] = rx; f7[4] = ry; f7[5] = rz;
        f7[6] = dis; f7[7] = 0.f;
    }
    __syncthreads();

    // aw raw = relu(x_rel . kernals + one_pad)
    for (int t = tid; t < TILE * K_ * KS_; t += 128) {
        int s = t % KS_; int pk = t / KS_; int k = pk % K_;
        const float* f7 = sF7 + pk * 8;
        float v = f7[3] * sKern[s] + f7[4] * sKern[KS_ + s] + f7[5] * sKern[2 * KS_ + s];
        if (k == 0 && s == 0) v += 1.0f;
        sAw[t] = v > 0.f ? v : 0.f;
    }
    __syncthreads();

    // per (p,s): normalize over k, square, normalize, threshold > 0.1
    for (int t = tid; t < TILE * KS_; t += 128) {
        int p = t / KS_, s = t % KS_;
        float* col = sAw + (p * K_) * KS_ + s;          // stride KS_
        float sum1 = 0.f;
#pragma unroll
        for (int k = 0; k < K_; ++k) sum1 += col[k * KS_];
        float r1 = 1.0f / (sum1 + 1e-6f);
        float sum2 = 0.f;
#pragma unroll
        for (int k = 0; k < K_; ++k) { float v = col[k * KS_] * r1; v = v * v; col[k * KS_] = v; sum2 += v; }
        float r2 = 1.0f / (sum2 + 1e-6f);
#pragma unroll
        for (int k = 0; k < K_; ++k) { float v = col[k * KS_] * r2; col[k * KS_] = v > 0.1f ? v : 0.f; }
    }
    __syncthreads();

    // agg[p][cc][s] = sum_k spirals[p][cc][k] * aw[p][k][s]
    const float* fb = feature + (size_t)b * C_ * N_;    // L2-resident (8MB)
    for (int t = tid; t < TILE * 2 * C_; t += 128) {
        int p = t / (2 * C_); int cc = t % (2 * C_);
        float spir[K_];
        if (cc < C_) {
            const float* fc = fb + (size_t)cc * N_;
            const int* ip = sIdx + p * K_;
#pragma unroll
            for (int k = 0; k < K_; ++k) spir[k] = fc[ip[k]];
        } else {
            int c2 = cc - C_;
            const float* w = sMW + c2 * 7;
            float bias = sMB[c2];
#pragma unroll
            for (int k = 0; k < K_; ++k) {
                const float* f7 = sF7 + (p * K_ + k) * 8;
                float acc = bias;
#pragma unroll
                for (int j = 0; j < 7; ++j) acc += w[j] * f7[j];
                spir[k] = acc;
            }
        }
        const float* awp = sAw + (p * K_) * KS_;
        float* ag = sAgg + p * 2048 + cc * KS_;
#pragma unroll
        for (int s = 0; s < KS_; ++s) {
            float acc = 0.f;
#pragma unroll
            for (int k = 0; k < K_; ++k) acc += spir[k] * awp[k * KS_ + s];
            ag[s] = acc;
        }
    }
    __syncthreads();

    // GEMM: out[16 pts, 64 oc] = sAgg[16,2048] @ conv_w.T  via V_WMMA_F32_16X16X4_F32
    // wave w (0..3) owns oc tile [16w, 16w+16). 512 WMMAs per wave over K=2048.
    int wave = tid >> 5, lane = tid & 31;
    if (wave < 4) {
        int ocb = wave * 16;
        int rc  = lane & 15;                // A: M row; B: N col (same lane mapping)
        int kk  = (lane >> 4) << 1;         // lanes 16-31 hold K+2,K+3
        v8f acc = {};
        const float* asrc = sAgg + rc * 2048 + kk;
        const float* bsrc = conv_w + (size_t)(ocb + rc) * 2048 + kk;
        for (int k = 0; k < 2048; k += 4) {
            v2f a  = *(const v2f*)(asrc + k);
            v2f bb = *(const v2f*)(bsrc + k);
            acc = __builtin_amdgcn_wmma_f32_16x16x4_f32(
                false, a, false, bb, (short)0, acc, false, false);
        }
        // C/D layout: VGPR r, lanes0-15 -> M=r, lanes16-31 -> M=r+8; N = lane&15
        int mofs = (lane >> 4) * 8;
        float cb = conv_b[ocb + rc];
        float* orow = outRaw + ((size_t)b * N_ + n0) * OC_;
#pragma unroll
        for (int r = 0; r < 8; ++r)
            orow[(size_t)(r + mofs) * OC_ + ocb + rc] = acc[r] + cb;
    }
}

// ---------------------------------------------------------------------------
// Kernel 3: BN stats per channel -> scale/shift
// ---------------------------------------------------------------------------
__global__ void bnstats_kernel(const float* __restrict__ outRaw,
                               const float* __restrict__ gamma,
                               const float* __restrict__ beta,
                               float* __restrict__ scale,
                               float* __restrict__ shift) {
    __shared__ float ssum[256], ssq[256];
    int oc = blockIdx.x;
    float sum = 0.f, sq = 0.f;
    for (int r = threadIdx.x; r < B_ * N_; r += 256) {
        float v = outRaw[(size_t)r * OC_ + oc];
        sum += v; sq += v * v;
    }
    ssum[threadIdx.x] = sum; ssq[threadIdx.x] = sq;
    __syncthreads();
    for (int st = 128; st > 0; st >>= 1) {
        if (threadIdx.x < st) {
            ssum[threadIdx.x] += ssum[threadIdx.x + st];
            ssq[threadIdx.x]  += ssq[threadIdx.x + st];
        }
        __syncthreads();
    }
    if (threadIdx.x == 0) {
        const float inv = 1.0f / (float)(B_ * N_);
        float mean = ssum[0] * inv;
        float var  = ssq[0] * inv - mean * mean;
        float sc = gamma[oc] * rsqrtf(var + 1e-5f);
        scale[oc] = sc;
        shift[oc] = beta[oc] - mean * sc;
    }
}

// ---------------------------------------------------------------------------
// Kernel 4: apply BN + transpose to [B, OC, N]
// ---------------------------------------------------------------------------
__global__ void bnapply_kernel(const float* __restrict__ outRaw,
                               const float* __restrict__ scale,
                               const float* __restrict__ shift,
                               float* __restrict__ out) {
    size_t t = (size_t)blockIdx.x * blockDim.x + threadIdx.x;
    if (t >= (size_t)B_ * OC_ * N_) return;
    int n  = (int)(t % N_);
    int oc = (int)((t / N_) % OC_);
    int b  = (int)(t / ((size_t)N_ * OC_));
    float v = outRaw[((size_t)b * N_ + n) * OC_ + oc];
    out[t] = v * scale[oc] + shift[oc];
}

// ---------------------------------------------------------------------------
extern "C" void kernel_launch(void* const* d_in, const int* in_sizes, int n_in,
                              void* d_out, int out_size, void* d_ws, size_t ws_size,
                              hipStream_t stream) {
    const float* x       = (const float*)d_in[0];
    const float* feature = (const float*)d_in[1];
    const float* kern    = (const float*)d_in[2];
    const float* mlp_w   = (const float*)d_in[3];
    const float* mlp_b   = (const float*)d_in[4];
    const float* conv_w  = (const float*)d_in[5];
    const float* conv_b  = (const float*)d_in[6];
    const float* gamma   = (const float*)d_in[7];
    const float* beta    = (const float*)d_in[8];

    char* ws = (char*)d_ws;
    int* idx = (int*)ws;                                         // 2.62 MB
    size_t off = (size_t)B_ * N_ * K_ * sizeof(int);
    float* outRaw = (float*)(ws + off);                          // 8.39 MB
    off += (size_t)B_ * N_ * OC_ * sizeof(float);
    float* scale = (float*)(ws + off); off += OC_ * sizeof(float);
    float* shift = (float*)(ws + off);

    knn_kernel<<<B_ * (N_ / 256), 256, 3 * N_ * sizeof(float), stream>>>(x, idx);

    size_t smemB = (size_t)(9520 + TILE * 2048) * sizeof(float); // ~169 KB of 320 KB WGP LDS
    paiconv_kernel<<<B_ * (N_ / TILE), 128, smemB, stream>>>(
        x, feature, kern, mlp_w, mlp_b, conv_w, conv_b, idx, outRaw);

    bnstats_kernel<<<OC_, 256, 0, stream>>>(outRaw, gamma, beta, scale, shift);

    int total = B_ * OC_ * N_;
    bnapply_kernel<<<(total + 255) / 256, 256, 0, stream>>>(outRaw, scale, shift, (float*)d_out);
}